// BiMambaBlock_14130442404376
// MI455X (gfx1250) — compile-verified
//
#include <hip/hip_runtime.h>

#define DMODEL 1024
#define DINNER 2048
#define DSTATE 16
#define DCONV  4
#define DTRANK 64
#define BSZ    4
#define SEQ    2048
#define NROW   (BSZ * SEQ)                 // 8192 rows (b*L)
#define PROJW  (DTRANK + 2 * DSTATE)       // 96

typedef __attribute__((ext_vector_type(2))) float v2f;
typedef __attribute__((ext_vector_type(8))) float v8f;

__device__ __forceinline__ int revrow(int r) {
  // flip time index within a batch (L = SEQ, power of two)
  return (r & ~(SEQ - 1)) | ((SEQ - 1) - (r & (SEQ - 1)));
}
__device__ __forceinline__ float softplusf(float v) {
  return v > 20.f ? v : log1pf(__expf(v));
}
__device__ __forceinline__ float siluf(float v) {
  return v / (1.f + __expf(-v));
}

// C[M,N] = act( A[M,K] @ W[N,K]^T + bias )   (optionally time-reversed reads/stores, accumulate)
// One wave -> 64x64 tile = 4x4 grid of 16x16 f32 WMMA tiles, K stepped by 4 with a
// 2-deep register pipeline: loads for step n+1 issue before the 16 WMMAs of step n.
// 32-bit lane offsets from uniform bases (SADDR-form loads); launch_bounds caps VGPRs
// so allocation stays under 256 (no MSB bank switching, >=4 waves/SIMD occupancy).
template <int ACT, int REV_IN, int REV_OUT, int ACCUM, int K, int LDA, int LDC>
__global__ __launch_bounds__(256, 4) void wmma_gemm_f32(
    const float* __restrict__ A,
    const float* __restrict__ W,
    const float* __restrict__ bias,
    float* __restrict__ C,
    int M, int N) {
  const int nT   = (N + 63) >> 6;
  const int wave = (blockIdx.x << 3) | (threadIdx.x >> 5);
  if (wave >= (M >> 6) * nT) return;           // wave-uniform exit (EXEC stays all-1 for WMMA)
  const int m64  = (wave / nT) << 6;
  const int n64  = (wave % nT) << 6;
  const int lane = threadIdx.x & 31;
  const int idx  = lane & 15;                  // M (A) / N (B) position within tile
  const int kb   = (lane >> 4) << 1;           // lanes 0-15: K=0,1 ; lanes 16-31: K=2,3

  int aoff[4];                                 // element offsets into A (fits 31 bits)
#pragma unroll
  for (int i = 0; i < 4; ++i) {
    int r = m64 + i * 16 + idx;
    if (REV_IN) r = revrow(r);
    aoff[i] = r * LDA + kb;
  }
  int woff[4];                                 // element offsets into W
#pragma unroll
  for (int j = 0; j < 4; ++j) {
    int n = n64 + j * 16 + idx;
    if (n > N - 1) n = N - 1;                  // dead tiles clamp; never stored
    woff[j] = n * K + kb;
  }

  const v8f zero8 = {0.f, 0.f, 0.f, 0.f, 0.f, 0.f, 0.f, 0.f};
  v8f acc[4][4];
#pragma unroll
  for (int i = 0; i < 4; ++i)
#pragma unroll
    for (int j = 0; j < 4; ++j) acc[i][j] = zero8;

  auto loadf = [&](v2f (&da)[4], v2f (&db)[4], int kk) {
#pragma unroll
    for (int i = 0; i < 4; ++i) da[i] = *(const v2f*)(A + (aoff[i] + kk));
#pragma unroll
    for (int j = 0; j < 4; ++j) db[j] = *(const v2f*)(W + (woff[j] + kk));
  };
  auto mma = [&](const v2f (&sa)[4], const v2f (&sb)[4]) {
#pragma unroll
    for (int i = 0; i < 4; ++i)
#pragma unroll
      for (int j = 0; j < 4; ++j)
        acc[i][j] = __builtin_amdgcn_wmma_f32_16x16x4_f32(
            false, sa[i], false, sb[j], (short)0, acc[i][j], false, false);
  };

  v2f a0[4], b0[4], a1[4], b1[4];
  loadf(a0, b0, 0);                      // prologue: k = 0
  for (int k0 = 0; k0 + 8 < K; k0 += 8) {
    loadf(a1, b1, k0 + 4);               // prefetch k+4
    mma(a0, b0);                         // compute k
    loadf(a0, b0, k0 + 8);               // prefetch k+8
    mma(a1, b1);                         // compute k+4
  }
  loadf(a1, b1, K - 4);                  // epilogue: last two steps
  mma(a0, b0);
  mma(a1, b1);

  // C/D layout: VGPR r, lanes 0-15 -> M=r, lanes 16-31 -> M=8+r ; N = lane&15
  const int mbase = (lane >> 4) << 3;
#pragma unroll
  for (int i = 0; i < 4; ++i) {
#pragma unroll
    for (int r = 0; r < 8; ++r) {
      int m = m64 + i * 16 + mbase + r;
      int mrow = REV_OUT ? revrow(m) : m;
#pragma unroll
      for (int j = 0; j < 4; ++j) {
        int n = n64 + j * 16 + idx;
        if (n < N) {
          float v = acc[i][j][r];
          if (ACT == 1) v = softplusf(v + bias[n]);
          size_t off = (size_t)mrow * LDC + n;
          if (ACCUM) C[off] += v;
          else       C[off] = v;
        }
      }
    }
  }
}

// xc[row,c] = silu( cb[c] + sum_k cw[c,k] * xi[b, l-3+k, c] ), xi = xz cols [0,2048)
// float4-vectorized over 4 consecutive channels.
__global__ __launch_bounds__(256) void conv_silu_kernel(
    const float* __restrict__ xz, const float* __restrict__ cw,
    const float* __restrict__ cb, float* __restrict__ xc) {
  int gid = blockIdx.x * 256 + threadIdx.x;        // NROW * DINNER/4 threads
  int row = gid >> 9;                              // / (DINNER/4)
  int cg  = gid & (DINNER / 4 - 1);                // channel group
  int c4  = cg << 2;
  int b   = row >> 11;                             // / SEQ
  int l   = row & (SEQ - 1);
  const float4* xz4 = (const float4*)xz;
  float acc[4];
#pragma unroll
  for (int u = 0; u < 4; ++u) acc[u] = cb[c4 + u];
#pragma unroll
  for (int k = 0; k < DCONV; ++k) {
    int l2 = l - (DCONV - 1) + k;
    if (l2 >= 0) {
      float4 v = xz4[(size_t)(b * SEQ + l2) * ((2 * DINNER) / 4) + cg];
      acc[0] = fmaf(cw[(c4 + 0) * DCONV + k], v.x, acc[0]);
      acc[1] = fmaf(cw[(c4 + 1) * DCONV + k], v.y, acc[1]);
      acc[2] = fmaf(cw[(c4 + 2) * DCONV + k], v.z, acc[2]);
      acc[3] = fmaf(cw[(c4 + 3) * DCONV + k], v.w, acc[3]);
    }
  }
  float4 o;
  o.x = siluf(acc[0]); o.y = siluf(acc[1]); o.z = siluf(acc[2]); o.w = siluf(acc[3]);
  ((float4*)xc)[(size_t)row * (DINNER / 4) + cg] = o;
}

// Selective scan: thread = (b, channel c); 16-state recurrence in registers.
// B_t/C_t (32 floats shared per (b,l)) staged through LDS in chunks of 32 steps.
// y written into xz cols [0,2048) (xi is dead after the conv).
__global__ __launch_bounds__(256) void scan_kernel(
    const float* __restrict__ dt, const float* __restrict__ xc,
    const float* __restrict__ xz, const float* __restrict__ proj,
    const float* __restrict__ A_log, const float* __restrict__ Dp,
    float* __restrict__ y) {
  const int b = blockIdx.x >> 3;
  const int c = ((blockIdx.x & 7) << 8) | threadIdx.x;
  float Ar[DSTATE];
#pragma unroll
  for (int s = 0; s < DSTATE; ++s) Ar[s] = -__expf(A_log[c * DSTATE + s]);
  const float Dc = Dp[c];
  float h[DSTATE];
#pragma unroll
  for (int s = 0; s < DSTATE; ++s) h[s] = 0.f;

  __shared__ float sBC[32][32];   // [t][j] j<16: B, j>=16: C
  for (int l0 = 0; l0 < SEQ; l0 += 32) {
    __syncthreads();
#pragma unroll
    for (int p = 0; p < 4; ++p) {
      int t = (p << 3) | (threadIdx.x >> 5);
      int j = threadIdx.x & 31;
      sBC[t][j] = proj[(size_t)(b * SEQ + l0 + t) * PROJW + DTRANK + j];
    }
    __syncthreads();
    for (int t = 0; t < 32; ++t) {
      const int row   = b * SEQ + l0 + t;
      const float dtv = dt[(size_t)row * DINNER + c];
      const float xv  = xc[(size_t)row * DINNER + c];
      const float zv  = xz[(size_t)row * (2 * DINNER) + DINNER + c];
      const float dtx = dtv * xv;
      float yv = 0.f;
#pragma unroll
      for (int s = 0; s < DSTATE; ++s) {
        float dA = __expf(dtv * Ar[s]);
        h[s] = fmaf(h[s], dA, dtx * sBC[t][s]);
        yv   = fmaf(h[s], sBC[t][DSTATE + s], yv);
      }
      yv = (yv + xv * Dc) * siluf(zv);
      y[(size_t)row * (2 * DINNER) + c] = yv;
    }
  }
}

extern "C" void kernel_launch(void* const* d_in, const int* in_sizes, int n_in,
                              void* d_out, int out_size, void* d_ws, size_t ws_size,
                              hipStream_t stream) {
  (void)in_sizes; (void)n_in; (void)out_size; (void)ws_size;
  const float* x    = (const float*)d_in[0];
  const float* in_w = (const float*)d_in[1];  // [2,4096,1024]
  const float* cw   = (const float*)d_in[2];  // [2,2048,4]
  const float* cb   = (const float*)d_in[3];  // [2,2048]
  const float* xw   = (const float*)d_in[4];  // [2,96,2048]
  const float* dtw  = (const float*)d_in[5];  // [2,2048,64]
  const float* dtb  = (const float*)d_in[6];  // [2,2048]
  const float* Alog = (const float*)d_in[7];  // [2,2048,16]
  const float* Dpar = (const float*)d_in[8];  // [2,2048]
  const float* ow   = (const float*)d_in[9];  // [2,1024,2048]
  float* out = (float*)d_out;

  float* xz   = (float*)d_ws;                           // 8192*4096  (xi | z); y reuses xi cols
  float* xc   = xz + (size_t)NROW * (2 * DINNER);       // 8192*2048
  float* proj = xc + (size_t)NROW * DINNER;             // 8192*96
  float* dt   = proj + (size_t)NROW * PROJW;            // 8192*2048

  auto gblocks = [](int M, int N) {
    int nT = (N + 63) / 64;
    int waves = (M / 64) * nT;
    return (waves + 7) / 8;                             // 8 waves (256 thr) per block
  };

  for (int d = 0; d < 2; ++d) {
    const float* inw_d = in_w + (size_t)d * (2 * DINNER) * DMODEL;
    const float* cw_d  = cw   + (size_t)d * DINNER * DCONV;
    const float* cb_d  = cb   + (size_t)d * DINNER;
    const float* xw_d  = xw   + (size_t)d * PROJW * DINNER;
    const float* dtw_d = dtw  + (size_t)d * DINNER * DTRANK;
    const float* dtb_d = dtb  + (size_t)d * DINNER;
    const float* Al_d  = Alog + (size_t)d * DINNER * DSTATE;
    const float* Dp_d  = Dpar + (size_t)d * DINNER;
    const float* ow_d  = ow   + (size_t)d * DMODEL * DINNER;

    // 1) in_proj: xz = (x or flip(x)) @ in_w^T
    if (d == 0)
      wmma_gemm_f32<0, 0, 0, 0, DMODEL, DMODEL, 2 * DINNER>
          <<<gblocks(NROW, 2 * DINNER), 256, 0, stream>>>(
              x, inw_d, nullptr, xz, NROW, 2 * DINNER);
    else
      wmma_gemm_f32<0, 1, 0, 0, DMODEL, DMODEL, 2 * DINNER>
          <<<gblocks(NROW, 2 * DINNER), 256, 0, stream>>>(
              x, inw_d, nullptr, xz, NROW, 2 * DINNER);

    // 2) depthwise causal conv + SiLU
    conv_silu_kernel<<<(NROW * (DINNER / 4)) / 256, 256, 0, stream>>>(xz, cw_d, cb_d, xc);

    // 3) x_proj: proj = xc @ xw^T   (N = 96; dead tiles clamp B rows, never stored)
    wmma_gemm_f32<0, 0, 0, 0, DINNER, DINNER, PROJW>
        <<<gblocks(NROW, PROJW), 256, 0, stream>>>(
            xc, xw_d, nullptr, proj, NROW, PROJW);

    // 4) dt = softplus(dt_raw @ dtw^T + dtb)   (dt_raw = proj cols [0,64))
    wmma_gemm_f32<1, 0, 0, 0, DTRANK, PROJW, DINNER>
        <<<gblocks(NROW, DINNER), 256, 0, stream>>>(
            proj, dtw_d, dtb_d, dt, NROW, DINNER);

    // 5) selective scan + D skip + SiLU(z) gate -> y (stored into xi cols of xz)
    scan_kernel<<<BSZ * (DINNER / 256), 256, 0, stream>>>(
        dt, xc, xz, proj, Al_d, Dp_d, xz);

    // 6) out_proj: out (+)= (flip?) y @ ow^T
    if (d == 0)
      wmma_gemm_f32<0, 0, 0, 0, DINNER, 2 * DINNER, DMODEL>
          <<<gblocks(NROW, DMODEL), 256, 0, stream>>>(
              xz, ow_d, nullptr, out, NROW, DMODEL);
    else
      wmma_gemm_f32<0, 0, 1, 1, DINNER, 2 * DINNER, DMODEL>
          <<<gblocks(NROW, DMODEL), 256, 0, stream>>>(
              xz, ow_d, nullptr, out, NROW, DMODEL);
  }
}